// DN1_83434034692102
// MI455X (gfx1250) — compile-verified
//
#include <hip/hip_runtime.h>
#include <math.h>

#define XD      4096
#define NN      8192
#define NZ      1024
#define NTOT    (NN + NZ)          // 9216
#define D       (XD + NN + NZ)     // 13312
#define TOPK    16
#define ZTOPK   8

#define KSPLIT        4
#define KCHUNK        (D / KSPLIT)         // 3328
#define ROWS_PER_WAVE 16
#define WAVES_PER_WG  8
#define ROWS_PER_WG   (ROWS_PER_WAVE * WAVES_PER_WG)  // 128
#define TILE_STRIDE   36                   // 16B-aligned pad, conflict-free

#define NEG_INF (-__builtin_inff())

typedef float v2f __attribute__((ext_vector_type(2)));
typedef float v8f __attribute__((ext_vector_type(8)));

// ---------------------------------------------------------------------------
// Kernel 0: inpt = concat(x, y_response, z)
// ---------------------------------------------------------------------------
__global__ __launch_bounds__(256) void k_build_inpt(
    const float* __restrict__ x, const float* __restrict__ y,
    const float* __restrict__ z, float* __restrict__ inpt) {
  int i = blockIdx.x * 256 + threadIdx.x;
  if (i < D) {
    float v;
    if (i < XD)            v = x[i];
    else if (i < XD + NN)  v = y[i - XD];
    else                   v = z[i - XD - NN];
    inpt[i] = v;
  }
}

// ---------------------------------------------------------------------------
// Kernel 1: fused neurons-copy + GEMV via V_WMMA_F32_16X16X4_F32.
// Each wave: 16 rows. Each block: 8 waves x (1/KSPLIT of K).
// B matrix is the inpt slice broadcast across all 16 columns, so
// C[m][0] accumulates response[row m] for this K-split.
// ---------------------------------------------------------------------------
__global__ __launch_bounds__(256) void k_gemv_copy(
    const float* __restrict__ Nin,
    const float* __restrict__ inpt_g,
    float* __restrict__ Nout,
    float* __restrict__ resp_part) {
  __shared__ float inpt_s[KCHUNK];
  __shared__ __align__(16) float tile[WAVES_PER_WG][ROWS_PER_WAVE * TILE_STRIDE];

  const int  tid    = threadIdx.x;
  const int  wave   = tid >> 5;
  const int  lane   = tid & 31;
  const int  split  = blockIdx.x & (KSPLIT - 1);
  const int  rowg   = blockIdx.x >> 2;
  const int  k_base = split * KCHUNK;
  const long row_base = (long)rowg * ROWS_PER_WG + (long)wave * ROWS_PER_WAVE;

  for (int i = tid; i < KCHUNK; i += 256) inpt_s[i] = inpt_g[k_base + i];
  __syncthreads();

  const int quad = lane >> 3;   // which 4-row group this lane loads
  const int c8   = lane & 7;    // which 16B column chunk
  const int l    = lane & 15;   // WMMA A: M row
  const int h    = lane >> 4;   // WMMA A/B: K half (K pairing 0,1 | 2,3)

  v8f c = {};
  float* mytile = tile[wave];

  for (int k0 = 0; k0 < KCHUNK; k0 += 32) {
    // 16 rows x 32 floats: coalesced 128B per 8 lanes; copy + stage to LDS.
#pragma unroll
    for (int rr = 0; rr < 4; ++rr) {
      const int  rl   = rr * 4 + quad;
      const long gidx = (row_base + rl) * (long)D + (long)(k_base + k0 + c8 * 4);
      float4 v = *(const float4*)(Nin + gidx);
      *(float4*)(Nout + gidx) = v;                         // output copy
      *(float4*)(mytile + rl * TILE_STRIDE + c8 * 4) = v;  // transpose staging
    }
    // 8 WMMAs consume the 16x32 tile, 4 K per WMMA.
#pragma unroll
    for (int koff = 0; koff < 32; koff += 4) {
      v2f a, b;
      a.x = mytile[l * TILE_STRIDE + koff + 2 * h];
      a.y = mytile[l * TILE_STRIDE + koff + 2 * h + 1];
      b.x = inpt_s[k0 + koff + 2 * h];
      b.y = inpt_s[k0 + koff + 2 * h + 1];
      c = __builtin_amdgcn_wmma_f32_16x16x4_f32(
          /*neg_a=*/false, a, /*neg_b=*/false, b,
          /*c_mod=*/(short)0, c, /*reuse_a=*/false, /*reuse_b=*/false);
    }
  }

  // C layout: VGPR j = row base+j (lanes 0-15, N=lane) / row base+8+j (lanes 16-31).
  float* rp = resp_part + (long)split * NTOT;
  if (lane == 0) {
#pragma unroll
    for (int j = 0; j < 8; ++j) rp[row_base + j] = c[j];
  } else if (lane == 16) {
#pragma unroll
    for (int j = 0; j < 8; ++j) rp[row_base + 8 + j] = c[j];
  }
}

// ---------------------------------------------------------------------------
// Kernel 2: sum K-split partials (fixed order -> deterministic), both top-k's
// (stable, lower index wins ties, matching lax.top_k), scatter final_rsp,
// copy ages + increment the 24 updated ones, emit update list.
// ---------------------------------------------------------------------------
__global__ __launch_bounds__(1024) void k_topk(
    const float* __restrict__ resp_part,
    const float* __restrict__ ages,
    float* __restrict__ out,
    int* __restrict__ upd_idx,
    float* __restrict__ upd_a) {
  __shared__ float yv[NN];
  __shared__ float zv[NZ];
  __shared__ float red_v[1024];
  __shared__ int   red_i[1024];
  __shared__ float s_zval[ZTOPK + 1];
  __shared__ int   s_zidx[ZTOPK + 1];
  __shared__ float s_yval[TOPK + 1];
  __shared__ int   s_yidx[TOPK + 1];
  __shared__ float s_denom;

  const int tid = threadIdx.x;
  float* out_z    = out;
  float* out_y    = out + NZ;
  float* out_ages = out + NZ + NN + (long)NTOT * D;

  for (int r = tid; r < NTOT; r += 1024) {
    float s = resp_part[r] + resp_part[NTOT + r] +
              resp_part[2 * NTOT + r] + resp_part[3 * NTOT + r];
    if (r < NN) yv[r] = s; else zv[r - NN] = s;
    out_ages[r] = ages[r];
  }
  for (int r = tid; r < NZ + NN; r += 1024) out[r] = 0.0f;  // final_rsp = 0
  __syncthreads();

  // ---- z top-(ZTOPK+1) over 1024 elements ----
  for (int k = 0; k < ZTOPK + 1; ++k) {
    red_v[tid] = zv[tid];
    red_i[tid] = tid;
    __syncthreads();
    for (int s = 512; s > 0; s >>= 1) {
      if (tid < s) {
        float v2 = red_v[tid + s]; int i2 = red_i[tid + s];
        if (v2 > red_v[tid] || (v2 == red_v[tid] && i2 < red_i[tid])) {
          red_v[tid] = v2; red_i[tid] = i2;
        }
      }
      __syncthreads();
    }
    if (tid == 0) { s_zval[k] = red_v[0]; s_zidx[k] = red_i[0]; zv[red_i[0]] = NEG_INF; }
    __syncthreads();
  }

  // ---- y top-(TOPK+1) over 8192 elements ----
  for (int k = 0; k < TOPK + 1; ++k) {
    float best = yv[tid]; int bi = tid;
    for (int j = tid + 1024; j < NN; j += 1024) {
      float v = yv[j];
      if (v > best) { best = v; bi = j; }  // ascending j: ties keep lower index
    }
    red_v[tid] = best; red_i[tid] = bi;
    __syncthreads();
    for (int s = 512; s > 0; s >>= 1) {
      if (tid < s) {
        float v2 = red_v[tid + s]; int i2 = red_i[tid + s];
        if (v2 > red_v[tid] || (v2 == red_v[tid] && i2 < red_i[tid])) {
          red_v[tid] = v2; red_i[tid] = i2;
        }
      }
      __syncthreads();
    }
    if (tid == 0) { s_yval[k] = red_v[0]; s_yidx[k] = red_i[0]; yv[red_i[0]] = NEG_INF; }
    __syncthreads();
  }

  if (tid == 0) {
    int t = 0;
    for (int i = 0; i < TOPK; ++i) if (s_yval[i] == s_yval[TOPK]) t = 1;
    s_denom = s_yval[0] - s_yval[TOPK] + 1e-9f * (t ? 0.5f : 0.0f);
  }
  __syncthreads();

  if (tid < ZTOPK) {
    float v = (s_zval[tid] - s_zval[ZTOPK]) / (s_zval[0] - s_zval[ZTOPK]);
    out_z[s_zidx[tid]] = v;
    int row = NN + s_zidx[tid];
    float a = ages[row];
    upd_idx[tid] = row;
    upd_a[tid]   = a;
    out_ages[row] = a + 1.0f;
  }
  if (tid < TOPK) {
    float v = (s_yval[tid] - s_yval[TOPK]) / s_denom;
    out_y[s_yidx[tid]] = v;
    int row = s_yidx[tid];
    float a = ages[row];
    upd_idx[ZTOPK + tid] = row;
    upd_a[ZTOPK + tid]   = a;
    out_ages[row] = a + 1.0f;
  }
}

// ---------------------------------------------------------------------------
// Kernel 3: update + renormalize 24 rows; one workgroup per row.
// ---------------------------------------------------------------------------
__global__ __launch_bounds__(256) void k_update(
    const float* __restrict__ Nin,
    const float* __restrict__ inpt,
    const int* __restrict__ upd_idx,
    const float* __restrict__ upd_a,
    float* __restrict__ Nout) {
  __shared__ float buf[D];
  __shared__ float red[256];
  const int   tid = threadIdx.x;
  const int   row = upd_idx[blockIdx.x];
  const float a   = upd_a[blockIdx.x];
  const float s   = (a - 1.0f) / a;
  const float inv = 1.0f / a;
  const float* src = Nin  + (long)row * D;
  float*       dst = Nout + (long)row * D;

  float local = 0.0f;
  for (int i = tid; i < D; i += 256) {
    float v = s * src[i] + inpt[i] * inv;
    buf[i] = v;
    local += v * v;
  }
  red[tid] = local;
  __syncthreads();
  for (int t = 128; t > 0; t >>= 1) {
    if (tid < t) red[tid] += red[tid + t];
    __syncthreads();
  }
  const float invn = 1.0f / sqrtf(red[0]);
  for (int i = tid; i < D; i += 256) dst[i] = buf[i] * invn;
}

// ---------------------------------------------------------------------------
extern "C" void kernel_launch(void* const* d_in, const int* in_sizes, int n_in,
                              void* d_out, int out_size, void* d_ws, size_t ws_size,
                              hipStream_t stream) {
  const float* x       = (const float*)d_in[0];
  const float* z       = (const float*)d_in[1];
  const float* neurons = (const float*)d_in[2];
  const float* ages    = (const float*)d_in[3];
  const float* y_resp  = (const float*)d_in[4];
  float* out = (float*)d_out;

  // ws layout (floats): [inpt D][resp_part KSPLIT*NTOT][upd_idx 32 (int, 24 used)][upd_a 24]
  float* wsf      = (float*)d_ws;
  float* ws_inpt  = wsf;
  float* ws_part  = wsf + D;
  int*   upd_idx  = (int*)(wsf + D + KSPLIT * NTOT);
  float* upd_a    = wsf + D + KSPLIT * NTOT + 32;

  float* out_neurons = out + NZ + NN;

  k_build_inpt<<<dim3((D + 255) / 256), dim3(256), 0, stream>>>(x, y_resp, z, ws_inpt);
  k_gemv_copy <<<dim3((NTOT / ROWS_PER_WG) * KSPLIT), dim3(256), 0, stream>>>(
      neurons, ws_inpt, out_neurons, ws_part);
  k_topk      <<<dim3(1), dim3(1024), 0, stream>>>(ws_part, ages, out, upd_idx, upd_a);
  k_update    <<<dim3(ZTOPK + TOPK), dim3(256), 0, stream>>>(
      neurons, ws_inpt, upd_idx, upd_a, out_neurons);
}